// DirectTuckerNet_67740224193009
// MI455X (gfx1250) — compile-verified
//
#include <hip/hip_runtime.h>

typedef __attribute__((ext_vector_type(2))) float v2f;
typedef __attribute__((ext_vector_type(4))) float v4f;
typedef __attribute__((ext_vector_type(8))) float v8f;

namespace {
constexpr int kI = 512;
constexpr int kJ = 512;
constexpr int kK = 512;
constexpr int kCols = kK * kJ;                                   // 262144 output columns
constexpr int kTilesPerWave = 8;                                 // 8 x 16-wide col tiles = 128 j per wave
constexpr int kWavesPerBlock = 8;                                // 256 threads
constexpr int kGroupsPerRowTile = kCols / (16 * kTilesPerWave);  // 2048 groups per 16-row tile
constexpr int kTotalWaves = (kI / 16) * kGroupsPerRowTile;       // 65536
constexpr int kBlocks = kTotalWaves / kWavesPerBlock;            // 8192
}

// out = M @ KR^T, M = A@G (512x16), KR[k*512+j, p*4+q] = C[k,p]*B[j,q]
// One wave -> eight 16x16 output tiles (16x128), each via 4x V_WMMA_F32_16X16X4_F32 (K = 16 total).
// k is constant per wave, so C[k,p] is folded into the A operand at setup;
// the B operand of all 4 chained WMMAs per tile is one directly-loaded b64.
__global__ __launch_bounds__(256, 1)
void tucker_wmma_kernel(const float* __restrict__ A,
                        const float* __restrict__ Bm,
                        const float* __restrict__ Cm,
                        const float* __restrict__ G,
                        float* __restrict__ out) {
  const int lane = (int)(threadIdx.x & 31u);
  const int wave = (int)(threadIdx.x >> 5u);
  const int w    = (int)blockIdx.x * kWavesPerBlock + wave;

  const int i_tile = w / kGroupsPerRowTile;
  const int g      = w % kGroupsPerRowTile;
  const int i0   = i_tile * 16;
  const int col0 = g * (16 * kTilesPerWave);   // 128-aligned -> single k for all 8 tiles
  const int k    = col0 / kJ;
  const int j0   = col0 % kJ;

  const int n  = lane & 15;   // output column within tile / A row within tile
  const int hi = lane >> 4;   // which K-half of the 16x4 / 4x16 operand this lane holds

  // ---- A operand: M-tile rows i0..i0+15, 16 cols (pq = p*4+q), as four 16x4
  // chunks, pre-scaled by C[k,c].
  // Lane layout (f32 A 16x4): lanes 0-15 row n K={0,1}; lanes 16-31 row n K={2,3}.
  v4f arow = *(const v4f*)(A + (size_t)(i0 + n) * 4);
  v4f crow = *(const v4f*)(Cm + (size_t)k * 4);   // uniform across wave
  v2f aop[4];
#pragma unroll
  for (int c = 0; c < 4; ++c) {
    const int pq = c * 4 + 2 * hi;
    v2f g0 = *(const v2f*)(G + 0 * 16 + pq);
    v2f g1 = *(const v2f*)(G + 1 * 16 + pq);
    v2f g2 = *(const v2f*)(G + 2 * 16 + pq);
    v2f g3 = *(const v2f*)(G + 3 * 16 + pq);
    v2f m = arow.x * g0 + arow.y * g1 + arow.z * g2 + arow.w * g3;
    aop[c] = m * crow[c];   // fold C[k,c] into A so B operand is chunk-invariant
  }

  // B operand base: lane grabs its 8-byte K'-slice directly (no selects).
  // Lanes 0-15: B[j0+n, 0..1]; lanes 16-31: B[j0+n, 2..3]. Per-tile step = 256 B (imm24).
  const float* bbase = Bm + (size_t)(j0 + n) * 4 + 2 * hi;

  // Store base: D layout = lanes 0-15 rows v (VGPR v), lanes 16-31 rows v+8, col = n.
  // All 64 store offsets (v*1MB + t*64B, max ~7.05MB) fit the signed imm24.
  float* obase = out + (size_t)(i0 + hi * 8) * kCols + (size_t)col0 + n;

#pragma unroll
  for (int t = 0; t < kTilesPerWave; ++t) {
    v2f bop = *(const v2f*)(bbase + t * (16 * 4));

    v8f acc = {};
#pragma unroll
    for (int c = 0; c < 4; ++c) {
      acc = __builtin_amdgcn_wmma_f32_16x16x4_f32(
          /*neg_a=*/false, aop[c], /*neg_b=*/false, bop,
          /*c_mod=*/(short)0, acc, /*reuse_a=*/false, /*reuse_b=*/false);
    }

#pragma unroll
    for (int v = 0; v < 8; ++v) {
      __builtin_nontemporal_store(acc[v], obase + (size_t)v * kCols + t * 16);
    }
  }
}

extern "C" void kernel_launch(void* const* d_in, const int* in_sizes, int n_in,
                              void* d_out, int out_size, void* d_ws, size_t ws_size,
                              hipStream_t stream) {
  const float* A = (const float*)d_in[0];
  const float* B = (const float*)d_in[1];
  const float* C = (const float*)d_in[2];
  const float* G = (const float*)d_in[3];
  float* out = (float*)d_out;

  hipLaunchKernelGGL(tucker_wmma_kernel, dim3(kBlocks), dim3(256), 0, stream,
                     A, B, C, G, out);
}